// spatial_strip_att_75453985457454
// MI455X (gfx1250) — compile-verified
//
#include <hip/hip_runtime.h>
#include <math.h>

// Fixed problem geometry from the reference
#define N_   8
#define C_   96
#define H_   256
#define W_   256
#define G_   2
#define K_   3
#define CG_  (C_ / G_)           // 48
#define PLANE (H_ * W_)          // 65536
#define NROWS (N_ * C_ * H_)     // 196608

typedef __attribute__((ext_vector_type(2))) float v2f;
typedef __attribute__((ext_vector_type(8))) float v8f;

// ---------------------------------------------------------------------------
// Kernel 1: per-(n,c) global average pool.  One block per plane (768 blocks).
// float4 coalesced loads; wave32 butterfly + small LDS cross-wave combine.
// ---------------------------------------------------------------------------
__global__ void gap_mean_kernel(const float* __restrict__ x,
                                float* __restrict__ m) {
    const int bid  = blockIdx.x;          // n*C_ + c
    const int t    = threadIdx.x;         // 0..255
    const int lane = t & 31;
    const int wv   = t >> 5;              // 0..7

    const float4* p = (const float4*)(x + (size_t)bid * PLANE);
    float s = 0.0f;
#pragma unroll 4
    for (int i = 0; i < 64; ++i) {        // 256 threads * 64 * 4 = 65536 floats
        float4 v = p[t + i * 256];
        s += (v.x + v.y) + (v.z + v.w);
    }
    // intra-wave reduction (wave32)
#pragma unroll
    for (int off = 16; off > 0; off >>= 1)
        s += __shfl_xor(s, off, 32);

    __shared__ float part[8];
    if (lane == 0) part[wv] = s;
    __syncthreads();
    if (t == 0) {
        float tot = 0.0f;
#pragma unroll
        for (int i = 0; i < 8; ++i) tot += part[i];
        m[bid] = tot * (1.0f / (float)PLANE);
    }
}

// ---------------------------------------------------------------------------
// Kernel 2: filt = tanh(m @ Wconv^T) via V_WMMA_F32_16X16X4_F32.
// One wave (32 threads); EXEC all-ones around the WMMA ops.
// A (16x4, rows = n, zero-padded past 8):  lane l holds row (l&15);
//   VGPR0 = K+ (l>=16 ? 2 : 0), VGPR1 = that +1   (ISA 32-bit A layout).
// B (4x16, cols = gk, zero-padded past 6): same K striping, col = l&15.
// D (16x16 f32, 8 VGPRs): lanes 0-15 hold M=v (v = VGPR idx) -> rows n=0..7.
// ---------------------------------------------------------------------------
__global__ void filt_wmma_kernel(const float* __restrict__ m,      // (8,96)
                                 const float* __restrict__ Wconv,  // (6,96)
                                 float* __restrict__ filt) {       // (8,6)
    const int l    = threadIdx.x;       // 0..31
    const int row  = l & 15;            // A row (n), valid < 8
    const int col  = l & 15;            // B col (gk), valid < 6
    const int koff = (l >= 16) ? 2 : 0;
    const int ar   = row & 7;           // clamped address row
    const int bc   = (col < 6) ? col : 0;
    const bool av  = (row < 8);
    const bool bv  = (col < 6);

    v8f acc = {};
    for (int kc = 0; kc < C_; kc += 4) {
        const int ka = kc + koff;
        v2f a, b;
        float a0 = m[ar * C_ + ka];
        float a1 = m[ar * C_ + ka + 1];
        float b0 = Wconv[bc * C_ + ka];
        float b1 = Wconv[bc * C_ + ka + 1];
        a.x = av ? a0 : 0.0f;
        a.y = av ? a1 : 0.0f;
        b.x = bv ? b0 : 0.0f;
        b.y = bv ? b1 : 0.0f;
        acc = __builtin_amdgcn_wmma_f32_16x16x4_f32(
            /*neg_a=*/false, a, /*neg_b=*/false, b,
            /*c_mod=*/(short)0, acc, /*reuse_a=*/false, /*reuse_b=*/false);
    }
    // D[v][l] for lanes 0..15 holds rows n = v = 0..7; write cols gk < 6
    if (l < 6) {
#pragma unroll
        for (int v = 0; v < 8; ++v)
            filt[v * (G_ * K_) + l] = tanhf(acc[v]);
    }
}

// ---------------------------------------------------------------------------
// Kernel 3: fused strip-conv + gap + affine mix.  One wave per 256-wide row,
// 8 rows per 256-thread block.  Lane holds 8 contiguous floats (two b128
// loads); taps via wave32 shuffles; row mean via __shfl_xor butterfly.
// ---------------------------------------------------------------------------
__global__ void strip_mix_kernel(const float* __restrict__ x,
                                 const float* __restrict__ filt,   // (8,2,3)
                                 const float* __restrict__ inside, // (96)
                                 const float* __restrict__ ll_p,   // (96)
                                 const float* __restrict__ lh_p,   // (96)
                                 float* __restrict__ out) {
    const int lane = threadIdx.x & 31;
    const int wv   = threadIdx.x >> 5;
    const int row  = blockIdx.x * 8 + wv;          // (n*C_ + c)*H_ + h
    const int nc   = row >> 8;                     // n*C_ + c  (H_ == 256)
    const int c    = nc % C_;
    const int n    = nc / C_;
    const int g    = (c >= CG_) ? 1 : 0;

    const float f0  = filt[(n * G_ + g) * K_ + 0];
    const float f1  = filt[(n * G_ + g) * K_ + 1];
    const float f2  = filt[(n * G_ + g) * K_ + 2];
    const float ia  = inside[c];
    const float ia1 = ia + 1.0f;
    const float ll  = ll_p[c];
    const float lh1 = lh_p[c] + 1.0f;

    const float4* xr = (const float4*)(x + (size_t)row * W_) + lane * 2;
    float4 v0 = xr[0];
    float4 v1 = xr[1];

    // prefetch the next block-stride of rows into L2 (global_prefetch_b8)
    if (row + 8 < NROWS)
        __builtin_prefetch(x + (size_t)(row + 8) * W_ + lane * 8, 0, 1);

    float xv[8] = {v0.x, v0.y, v0.z, v0.w, v1.x, v1.y, v1.z, v1.w};

    // neighbors across lanes; reflect pad at row edges: x[-1]=x[1], x[256]=x[254]
    float left  = __shfl_up(v1.w, 1, 32);
    float right = __shfl_down(v0.x, 1, 32);
    if (lane == 0)  left  = v0.y;   // x[1]
    if (lane == 31) right = v1.z;   // x[254]

    // row mean (gap), wave32 butterfly
    float s = 0.0f;
#pragma unroll
    for (int j = 0; j < 8; ++j) s += xv[j];
#pragma unroll
    for (int off = 16; off > 0; off >>= 1)
        s += __shfl_xor(s, off, 32);
    const float gap = s * (1.0f / (float)W_);
    const float gterm = ia * gap;

    float4 o0, o1;
    float o[8];
#pragma unroll
    for (int j = 0; j < 8; ++j) {
        const float xm = (j == 0) ? left  : xv[j - 1];
        const float xp = (j == 7) ? right : xv[j + 1];
        const float conv = fmaf(xm, f0, fmaf(xv[j], f1, xp * f2));
        o[j] = fmaf(fmaf(conv, ia1, -gterm), ll, xv[j] * lh1);
    }
    o0.x = o[0]; o0.y = o[1]; o0.z = o[2]; o0.w = o[3];
    o1.x = o[4]; o1.y = o[5]; o1.z = o[6]; o1.w = o[7];

    float4* orow = (float4*)(out + (size_t)row * W_) + lane * 2;
    orow[0] = o0;
    orow[1] = o1;
}

// ---------------------------------------------------------------------------
extern "C" void kernel_launch(void* const* d_in, const int* in_sizes, int n_in,
                              void* d_out, int out_size, void* d_ws, size_t ws_size,
                              hipStream_t stream) {
    const float* x      = (const float*)d_in[0];
    const float* Wconv  = (const float*)d_in[1];
    const float* inside = (const float*)d_in[2];
    const float* lamb_l = (const float*)d_in[3];
    const float* lamb_h = (const float*)d_in[4];
    float* out = (float*)d_out;

    float* m    = (float*)d_ws;          // 768 floats
    float* filt = m + N_ * C_;           // 48 floats

    gap_mean_kernel<<<N_ * C_, 256, 0, stream>>>(x, m);
    filt_wmma_kernel<<<1, 32, 0, stream>>>(m, Wconv, filt);
    strip_mix_kernel<<<NROWS / 8, 256, 0, stream>>>(x, filt, inside, lamb_l,
                                                    lamb_h, out);
}